// HyperConv_23871428231485
// MI455X (gfx1250) — compile-verified
//
#include <hip/hip_runtime.h>

// HyperConv on MI455X (gfx1250): 3x COO SpMM + accumulate + average.
// ~2 GFLOP vs ~115MB working set (all L2-resident, 192MB L2): bound by L2
// bandwidth + scatter-atomic throughput, not compute. WMMA is inapplicable
// (per-edge random gather/scatter); CDNA5 paths used instead:
//   global_load_async_to_lds_b128 + s_wait_asynccnt (edge-list staging DMA)
//   global_prefetch_b8 (gather-row prefetch)
//   b128 vector gathers (one wave32 covers a 512B embedding row)
//   non-returning global_atomic_add_f32 (relaxed, agent scope)

#define EMB     128
#define THREADS 256
#define EPB     2048                  // edges staged per block (24 KB LDS)
#define EPW     (EPB / (THREADS / 32))// 256 edges per wave32

#if __has_builtin(__builtin_amdgcn_global_load_async_to_lds_b128) && \
    __has_builtin(__builtin_amdgcn_s_wait_asynccnt)
#define ASYNC_LDS 1
#else
#define ASYNC_LDS 0
#endif

#if ASYNC_LDS
typedef int v4i __attribute__((ext_vector_type(4)));
#define AS1V(p) ((__attribute__((address_space(1))) v4i*)(p))
#define AS3V(p) ((__attribute__((address_space(3))) v4i*)(p))
#endif

__device__ __forceinline__ void atomic_add_f32_noret(float* p, float v)
{
    // Relaxed, agent-scope, result discarded -> non-returning global_atomic_add_f32.
    (void)__hip_atomic_fetch_add(p, v, __ATOMIC_RELAXED, __HIP_MEMORY_SCOPE_AGENT);
}

// ---------------------------------------------------------------------------
// ISA probe (never launched): placed first in the object so the disasm snippet
// shows the exact lowering of the async-LDS DMA and the f32 atomic add.
// ---------------------------------------------------------------------------
__global__ void hc_isa_probe(const int* __restrict__ g, float* __restrict__ f)
{
    __shared__ int s[64];
#if ASYNC_LDS
    __builtin_amdgcn_global_load_async_to_lds_b128(AS1V(g + threadIdx.x * 4),
                                                   AS3V(&s[(threadIdx.x & 15) * 4]), 0, 0);
    __builtin_amdgcn_s_wait_asynccnt(0);
#else
    s[threadIdx.x & 63] = g[threadIdx.x];
#endif
    __syncthreads();
    atomic_add_f32_noret(f + s[threadIdx.x & 63], 1.0f);
}

__global__ __launch_bounds__(THREADS)
void hc_spmm_scatter(const int* __restrict__ rows, const int* __restrict__ cols,
                     const float* __restrict__ vals, const float* __restrict__ cur,
                     float* __restrict__ nxt, int nnz)
{
    __shared__ int   s_rows[EPB];
    __shared__ int   s_cols[EPB];
    __shared__ float s_vals[EPB];

    const int tid  = threadIdx.x;
    const int base = blockIdx.x * EPB;
    const bool full = (base + EPB) <= nnz;   // uniform: all blocks but the last

    if (full) {
        // Guard-free staging: 2 b128 groups per thread per array (6 async ops).
#pragma unroll
        for (int k = 0; k < 2; ++k) {
            const int e0 = (tid + k * THREADS) * 4;
            const int g  = base + e0;
#if ASYNC_LDS
            __builtin_amdgcn_global_load_async_to_lds_b128(AS1V(rows + g), AS3V(&s_rows[e0]), 0, 0);
            __builtin_amdgcn_global_load_async_to_lds_b128(AS1V(cols + g), AS3V(&s_cols[e0]), 0, 0);
            __builtin_amdgcn_global_load_async_to_lds_b128(AS1V(vals + g), AS3V(&s_vals[e0]), 0, 0);
#else
            *(int4*)  &s_rows[e0] = *(const int4*)  (rows + g);
            *(int4*)  &s_cols[e0] = *(const int4*)  (cols + g);
            *(float4*)&s_vals[e0] = *(const float4*)(vals + g);
#endif
        }
    } else {
        // Ragged last block: scalar staging (DS stores), bounds-checked.
        for (int e0 = tid; e0 < EPB; e0 += THREADS) {
            const int g = base + e0;
            if (g < nnz) {
                s_rows[e0] = rows[g];
                s_cols[e0] = cols[g];
                s_vals[e0] = vals[g];
            }
        }
    }
#if ASYNC_LDS
    __builtin_amdgcn_s_wait_asynccnt(0);
#endif
    __syncthreads();

    const int wave = tid >> 5;
    const int lane = tid & 31;
    const int eBeg = wave * EPW;

    // Per-wave trip count (uniform within the wave; no per-iteration checks).
    int rem = nnz - base - eBeg;
    const int limit = rem < 0 ? 0 : (rem > EPW ? EPW : rem);

    // One wave per edge iteration: 32 lanes x float4 cover the 128-wide row.
    for (int i = 0; i < limit; ++i) {
        const int e = eBeg + i;

        // Prefetch a future gather row (4 lanes cover the 512B row).
        if (((i & 7) == 0) && lane < 4) {
            const int ep = e + 32;
            if (ep < EPB && (base + ep) < nnz)
                __builtin_prefetch(cur + (size_t)s_cols[ep] * EMB + lane * 32, 0, 1);
        }

        const int   r = s_rows[e];     // LDS broadcast reads
        const int   c = s_cols[e];
        const float v = s_vals[e];

        const float4 g = ((const float4*)(cur + (size_t)c * EMB))[lane]; // b128 gather
        float* dst = nxt + (size_t)r * EMB + lane * 4;
        atomic_add_f32_noret(dst + 0, g.x * v);
        atomic_add_f32_noret(dst + 1, g.y * v);
        atomic_add_f32_noret(dst + 2, g.z * v);
        atomic_add_f32_noret(dst + 3, g.w * v);
    }
}

__global__ __launch_bounds__(THREADS)
void hc_init(const float4* __restrict__ t, float4* __restrict__ acc,
             float4* __restrict__ cur, int n4)
{
    const int i = blockIdx.x * blockDim.x + threadIdx.x;
    if (i < n4) { const float4 v = t[i]; acc[i] = v; cur[i] = v; }
}

__global__ __launch_bounds__(THREADS)
void hc_zero(float4* __restrict__ p, int n4)
{
    const int i = blockIdx.x * blockDim.x + threadIdx.x;
    if (i < n4) p[i] = make_float4(0.f, 0.f, 0.f, 0.f);
}

__global__ __launch_bounds__(THREADS)
void hc_add(float4* __restrict__ acc, const float4* __restrict__ x, int n4)
{
    const int i = blockIdx.x * blockDim.x + threadIdx.x;
    if (i < n4) {
        float4 a = acc[i]; const float4 b = x[i];
        a.x += b.x; a.y += b.y; a.z += b.z; a.w += b.w;
        acc[i] = a;
    }
}

__global__ __launch_bounds__(THREADS)
void hc_add_scale(float4* __restrict__ acc, const float4* __restrict__ x,
                  float s, int n4)
{
    const int i = blockIdx.x * blockDim.x + threadIdx.x;
    if (i < n4) {
        float4 a = acc[i]; const float4 b = x[i];
        a.x = (a.x + b.x) * s; a.y = (a.y + b.y) * s;
        a.z = (a.z + b.z) * s; a.w = (a.w + b.w) * s;
        acc[i] = a;
    }
}

extern "C" void kernel_launch(void* const* d_in, const int* in_sizes, int n_in,
                              void* d_out, int out_size, void* d_ws, size_t ws_size,
                              hipStream_t stream)
{
    const int*   rows  = (const int*)  d_in[0];
    const int*   cols  = (const int*)  d_in[1];
    const float* vals  = (const float*)d_in[2];
    const float* table = (const float*)d_in[3];

    const int nnz   = in_sizes[0];
    const int nElem = in_sizes[3];     // N_ITEMS * 128
    const int n4    = nElem / 4;

    float* acc = (float*)d_out;
    float* cur = (float*)d_ws;         // ping-pong buffers in workspace
    float* nxt = cur + (size_t)nElem;  // needs 2*nElem*4 bytes (~42 MB)

    const int eb = (n4 + THREADS - 1) / THREADS;
    const int sb = (nnz + EPB - 1) / EPB;

    hc_init<<<eb, THREADS, 0, stream>>>((const float4*)table, (float4*)acc,
                                        (float4*)cur, n4);

    for (int layer = 0; layer < 3; ++layer) {
        hc_zero<<<eb, THREADS, 0, stream>>>((float4*)nxt, n4);
        hc_spmm_scatter<<<sb, THREADS, 0, stream>>>(rows, cols, vals, cur, nxt, nnz);
        if (layer < 2)
            hc_add<<<eb, THREADS, 0, stream>>>((float4*)acc, (const float4*)nxt, n4);
        else
            hc_add_scale<<<eb, THREADS, 0, stream>>>((float4*)acc, (const float4*)nxt,
                                                     0.25f, n4);
        float* t = cur; cur = nxt; nxt = t;
    }
}